// SelectiveScanPurePyTorch_34368328303315
// MI455X (gfx1250) — compile-verified
//
#include <hip/hip_runtime.h>
#include <hip/hip_bf16.h>

// ---------------- problem geometry (fixed by the reference) ----------------
#define D_MODEL 64
#define D_STATE 16
#define D_INNER 128
#define NB      4
#define NH      128
#define NW      64
#define NL      (NH * NW)      // 8192
#define NROWS   (NB * NL)      // 32768 total "pixels"
#define XDBL_LD 48             // 33 cols (dt,B,C) padded to 48 for 16-wide tiles
#define CHUNK   128
#define NCHUNK  (NL / CHUNK)   // 64 chunks per batch image

typedef __attribute__((ext_vector_type(2))) float v2f;
typedef __attribute__((ext_vector_type(8))) float v8f;

__device__ __forceinline__ float dev_sigmoid(float v) {
    return 1.0f / (1.0f + __expf(-v));
}
__device__ __forceinline__ float dev_silu(float v) {
    return v * dev_sigmoid(v);
}
__device__ __forceinline__ float dev_softplus(float v) {
    return (v > 20.0f) ? v : __logf(1.0f + __expf(v));
}

// ---------------------------------------------------------------------------
// Generic fp32 WMMA GEMM:  out[r, n] = sum_k A[r,k] * Wt[n,k]
//   one wave per block; wave owns a 16-row x (16*NT)-col tile of out.
//   All shape/layout switches are compile-time so the k-loop fully unrolls
//   into immediate-offset global_load clauses + v_wmma_f32_16x16x4_f32.
//   AMODE  0: A row-major, leading dim LDA
//   AMODE  1: A is x in (B, K, NL) layout: A[r,k] = x[(r>>13)*K*NL + k*NL + (r&8191)]
//   STOREM 0: out[r*ldo + n]
//   STOREM 1: out[((r>>13)*nTot + n)*NL + (r&8191)]   (final NCHW store)
//   MASK: guard weight rows n >= nValid (x_proj, N=33) via select (EXEC stays full)
// ---------------------------------------------------------------------------
template <int NT, int K, int LDA, int AMODE, int STOREM, bool MASK>
__global__ void __launch_bounds__(32)
k_gemm_wmma_f32(const float* __restrict__ A,
                const float* __restrict__ Wt, int nValid,
                float* __restrict__ out, int ldo, int nTot) {
    const int lane    = threadIdx.x;        // 0..31
    const int rowBase = blockIdx.x * 16;
    const int nBase   = blockIdx.y * (16 * NT);
    const int rA   = lane & 15;             // A row within tile
    const int kHi  = (lane >> 4) * 2;       // lanes 0-15 -> k,k+1 ; 16-31 -> k+2,k+3
    const int nCol = lane & 15;

    // warm the (tiny, heavily reused) weight panel into cache
    __builtin_prefetch(Wt + (size_t)(nBase + nCol) * K, 0, 3);

    // per-lane base pointers; k advances by immediate offsets after unroll
    const float* aBase;
    if (AMODE == 0) {
        aBase = A + (size_t)(rowBase + rA) * LDA + kHi;
    } else {
        const int r = rowBase + rA;
        aBase = A + (size_t)(r >> 13) * ((size_t)K * NL) + (size_t)(r & (NL - 1))
                  + (size_t)kHi * NL;
    }
    const float* wBase[NT];
    float msk[NT];
#pragma unroll
    for (int t = 0; t < NT; ++t) {
        const int n = nBase + t * 16 + nCol;
        const int nc = MASK ? ((n < nValid) ? n : (nValid - 1)) : n;
        msk[t] = MASK ? ((n < nValid) ? 1.0f : 0.0f) : 1.0f;
        wBase[t] = Wt + (size_t)nc * K + kHi;
    }

    v8f acc[NT];
#pragma unroll
    for (int t = 0; t < NT; ++t) acc[t] = v8f{0.f,0.f,0.f,0.f,0.f,0.f,0.f,0.f};

#pragma unroll
    for (int kb = 0; kb < K; kb += 4) {
        v2f a;
        if (AMODE == 0) {
            a.x = aBase[kb];
            a.y = aBase[kb + 1];
        } else {
            a.x = aBase[(size_t)kb * NL];
            a.y = aBase[(size_t)(kb + 1) * NL];
        }
#pragma unroll
        for (int t = 0; t < NT; ++t) {
            v2f bm;
            bm.x = wBase[t][kb];
            bm.y = wBase[t][kb + 1];
            if (MASK) { bm.x *= msk[t]; bm.y *= msk[t]; }
            acc[t] = __builtin_amdgcn_wmma_f32_16x16x4_f32(
                false, a, false, bm, (short)0, acc[t], false, false);
        }
    }

    // D layout: lane, vgpr v  ->  M = v + 8*(lane>>4), N = lane&15
    const int mOff = 8 * (lane >> 4);
#pragma unroll
    for (int t = 0; t < NT; ++t) {
        const int n = nBase + t * 16 + nCol;
#pragma unroll
        for (int v = 0; v < 8; ++v) {
            const int rr = rowBase + mOff + v;
            if (STOREM == 0) {
                out[(size_t)rr * ldo + n] = acc[t][v];
            } else {
                const int b = rr >> 13;
                const int l = rr & (NL - 1);
                out[((size_t)b * nTot + n) * NL + l] = acc[t][v];
            }
        }
    }
}

// ---------------------------------------------------------------------------
// Depthwise 3x3 conv (pad 1) + bias + SiLU.
// reads xs = cols [0,128) of the xz buffer (B, NL, 256); writes u (B, NL, 128).
// ---------------------------------------------------------------------------
__global__ void k_dwconv_silu(const float* __restrict__ xz,
                              const float* __restrict__ cw,
                              const float* __restrict__ cb,
                              float* __restrict__ u) {
    const int idx = blockIdx.x * blockDim.x + threadIdx.x;
    if (idx >= NB * NL * D_INNER) return;
    const int d  = idx & (D_INNER - 1);
    const int l  = (idx >> 7) & (NL - 1);
    const int b  = idx >> 20;
    const int hh = l >> 6;   // / NW
    const int ww = l & (NW - 1);

    float acc = cb[d];
#pragma unroll
    for (int kh = 0; kh < 3; ++kh) {
        const int h2 = hh + kh - 1;
        if ((unsigned)h2 >= (unsigned)NH) continue;
#pragma unroll
        for (int kw = 0; kw < 3; ++kw) {
            const int w2 = ww + kw - 1;
            if ((unsigned)w2 >= (unsigned)NW) continue;
            acc += xz[((size_t)b * NL + (size_t)h2 * NW + w2) * 256 + d]
                 * cw[d * 9 + kh * 3 + kw];
        }
    }
    u[idx] = dev_silu(acc);
}

// ---------------------------------------------------------------------------
// Selective scan, chunked two-phase.
// Thread = one (b, d); 16 states live in registers.
// xdbl row: [0]=dt_in, [1..16]=B, [17..32]=C (cols 33..47 padding).
// ---------------------------------------------------------------------------
__global__ void __launch_bounds__(128)
k_scan_phase1(const float* __restrict__ xdbl, const float* __restrict__ u,
              const float* __restrict__ A_log, const float* __restrict__ dtw,
              const float* __restrict__ dtb,
              float* __restrict__ Pbuf, float* __restrict__ Sbuf) {
    const int d     = threadIdx.x;
    const int chunk = blockIdx.x;
    const int b     = blockIdx.y;

    float Arow[D_STATE];
#pragma unroll
    for (int n = 0; n < D_STATE; ++n) Arow[n] = -__expf(A_log[d * D_STATE + n]);
    const float w  = dtw[d];
    const float bb = dtb[d];

    float P[D_STATE], S[D_STATE];
#pragma unroll
    for (int n = 0; n < D_STATE; ++n) { P[n] = 1.0f; S[n] = 0.0f; }

    const int l0 = chunk * CHUNK;
    for (int t = 0; t < CHUNK; ++t) {
        const int l = l0 + t;
        const float* xr = xdbl + (size_t)(b * NL + l) * XDBL_LD;
        const float dtin = xr[0];
        const float uval = u[(size_t)(b * NL + l) * D_INNER + d];
        const float dtv  = dev_softplus(dtin * w + bb);
        const float du   = dtv * uval;
#pragma unroll
        for (int n = 0; n < D_STATE; ++n) {
            const float dA = __expf(dtv * Arow[n]);
            S[n] = dA * S[n] + du * xr[1 + n];
            P[n] *= dA;
        }
    }
    const size_t base = ((size_t)((b * NCHUNK + chunk) * D_INNER + d)) * D_STATE;
#pragma unroll
    for (int n = 0; n < D_STATE; ++n) { Pbuf[base + n] = P[n]; Sbuf[base + n] = S[n]; }
}

__global__ void __launch_bounds__(128)
k_scan_combine(const float* __restrict__ Pbuf, const float* __restrict__ Sbuf,
               float* __restrict__ H0) {
    const int d = threadIdx.x;
    const int b = blockIdx.x;
    float h[D_STATE];
#pragma unroll
    for (int n = 0; n < D_STATE; ++n) h[n] = 0.0f;
    for (int c = 0; c < NCHUNK; ++c) {
        const size_t base = ((size_t)((b * NCHUNK + c) * D_INNER + d)) * D_STATE;
#pragma unroll
        for (int n = 0; n < D_STATE; ++n) {
            H0[base + n] = h[n];                       // state entering this chunk
            h[n] = Pbuf[base + n] * h[n] + Sbuf[base + n];
        }
    }
}

__global__ void __launch_bounds__(128)
k_scan_phase2(const float* __restrict__ xdbl, const float* __restrict__ u,
              const float* __restrict__ xz,
              const float* __restrict__ A_log, const float* __restrict__ dtw,
              const float* __restrict__ dtb, const float* __restrict__ Dp,
              const float* __restrict__ H0, float* __restrict__ yb) {
    const int d     = threadIdx.x;
    const int chunk = blockIdx.x;
    const int b     = blockIdx.y;

    float Arow[D_STATE];
#pragma unroll
    for (int n = 0; n < D_STATE; ++n) Arow[n] = -__expf(A_log[d * D_STATE + n]);
    const float w  = dtw[d];
    const float bb = dtb[d];
    const float Dd = Dp[d];

    float h[D_STATE];
    {
        const size_t base = ((size_t)((b * NCHUNK + chunk) * D_INNER + d)) * D_STATE;
#pragma unroll
        for (int n = 0; n < D_STATE; ++n) h[n] = H0[base + n];
    }

    const int l0 = chunk * CHUNK;
    for (int t = 0; t < CHUNK; ++t) {
        const int l = l0 + t;
        const float* xr = xdbl + (size_t)(b * NL + l) * XDBL_LD;
        const float dtin = xr[0];
        const float uval = u[(size_t)(b * NL + l) * D_INNER + d];
        const float dtv  = dev_softplus(dtin * w + bb);
        const float du   = dtv * uval;
        float y = uval * Dd;
#pragma unroll
        for (int n = 0; n < D_STATE; ++n) {
            const float dA = __expf(dtv * Arow[n]);
            h[n] = dA * h[n] + du * xr[1 + n];
            y += h[n] * xr[17 + n];
        }
        // source gating quirk: zf[b,l,d] = z[b, h=d, w=l>>7, c=l&127],
        // z stored as cols [128,256) of xz (B, NL, 256) channel-last.
        const float zv = xz[((size_t)b * NL + (size_t)d * NW + (l >> 7)) * 256
                            + 128 + (l & 127)];
        y *= dev_silu(zv);
        yb[(size_t)(b * NL + l) * D_INNER + d] = y;
    }
}

// ---------------------------------------------------------------------------
extern "C" void kernel_launch(void* const* d_in, const int* in_sizes, int n_in,
                              void* d_out, int out_size, void* d_ws, size_t ws_size,
                              hipStream_t stream) {
    const float* x    = (const float*)d_in[0];   // (4, 64, 128, 64)
    const float* ipw  = (const float*)d_in[1];   // (256, 64)
    const float* cw   = (const float*)d_in[2];   // (128, 1, 3, 3)
    const float* cb   = (const float*)d_in[3];   // (128,)
    const float* xpw  = (const float*)d_in[4];   // (33, 128)
    const float* dtw  = (const float*)d_in[5];   // (128, 1)
    const float* dtb  = (const float*)d_in[6];   // (128,)
    const float* Alog = (const float*)d_in[7];   // (128, 16)
    const float* Dp   = (const float*)d_in[8];   // (128,)
    const float* opw  = (const float*)d_in[9];   // (64, 128)
    float* out = (float*)d_out;                  // (4, 64, 128, 64)

    // workspace carve-up (floats); all regions fully written before read
    float* ws   = (float*)d_ws;
    float* xz   = ws;                               // (B, NL, 256)   8.4M
    float* u    = xz   + (size_t)NROWS * 256;       // (B, NL, 128)   4.2M
    float* xdbl = u    + (size_t)NROWS * D_INNER;   // (B, NL, 48)    1.6M
    float* Pbuf = xdbl + (size_t)NROWS * XDBL_LD;   // (B,64,128,16)  0.5M
    float* Sbuf = Pbuf + (size_t)NB * NCHUNK * D_INNER * D_STATE;
    float* H0   = Sbuf + (size_t)NB * NCHUNK * D_INNER * D_STATE;
    float* yb   = H0   + (size_t)NB * NCHUNK * D_INNER * D_STATE;  // (B, NL, 128)

    // 1) in_proj: xz[r, o] = sum_c x[b,c,l] * ipw[o,c]   (K=64, N=256)
    k_gemm_wmma_f32<8, D_MODEL, 0, /*AMODE=*/1, /*STOREM=*/0, false>
        <<<dim3(NROWS / 16, 256 / (16 * 8)), 32, 0, stream>>>(
            x, ipw, 256, xz, 256, 256);

    // 2) depthwise conv + SiLU -> u
    k_dwconv_silu<<<(NROWS * D_INNER) / 256, 256, 0, stream>>>(xz, cw, cb, u);

    // 3) x_proj: xdbl[r, n] = sum_d u[r,d] * xpw[n,d]    (K=128, N=33 pad 48)
    k_gemm_wmma_f32<3, D_INNER, D_INNER, /*AMODE=*/0, /*STOREM=*/0, true>
        <<<dim3(NROWS / 16, 1), 32, 0, stream>>>(
            u, xpw, 33, xdbl, XDBL_LD, XDBL_LD);

    // 4) chunked selective scan
    k_scan_phase1<<<dim3(NCHUNK, NB), D_INNER, 0, stream>>>(
        xdbl, u, Alog, dtw, dtb, Pbuf, Sbuf);
    k_scan_combine<<<NB, D_INNER, 0, stream>>>(Pbuf, Sbuf, H0);
    k_scan_phase2<<<dim3(NCHUNK, NB), D_INNER, 0, stream>>>(
        xdbl, u, xz, Alog, dtw, dtb, Dp, H0, yb);

    // 5) out_proj with fused NCHW transpose store (K=128, N=64)
    k_gemm_wmma_f32<4, D_INNER, D_INNER, /*AMODE=*/0, /*STOREM=*/1, false>
        <<<dim3(NROWS / 16, 1), 32, 0, stream>>>(
            yb, opw, 64, out, 64, 64);
}